// DetectionConfidenceMap2keypoint_39926015984318
// MI455X (gfx1250) — compile-verified
//
#include <hip/hip_runtime.h>
#include <hip/hip_bf16.h>

typedef float v2f __attribute__((ext_vector_type(2)));
typedef float v8f __attribute__((ext_vector_type(8)));

#define HH 128
#define WW 128
#define MAPELEMS (HH * WW)            // 16384
#define NMAPS    1024                 // B*C = 16*64
#define OFF_MAP  0
#define OFF_KP   16777216             // B*C*H*W
#define OFF_ZETA (OFF_KP + 2048)      // + B*C*2
#define OFF_TFKP (OFF_ZETA + 1024)    // + B*C

// One wave32 per (b,c) map. Reductions run on the matrix pipe with exact-fp32
// V_WMMA_F32_16X16X4_F32:
//   D0 += ones(16x4)   x B(4x16 of |R|)  -> per-column sums
//   D1 += rowidx(16x4) x B               -> per-column sum_i i*|R|
// STORE_MAP=true additionally writes |R| (map_val output) for the first tensor.
template <bool STORE_MAP>
__global__ __launch_bounds__(128)
void soft_argmax_wmma_kernel(const float* __restrict__ src,
                             float* __restrict__ out,
                             int kp_off)   // OFF_KP or OFF_TFKP
{
    const int wave = threadIdx.x >> 5;
    const int lane = threadIdx.x & 31;
    const int m    = blockIdx.x * 4 + wave;   // map index = b*C + c, 0..1023

    const float* __restrict__ p = src + (size_t)m * MAPELEMS;
    float* __restrict__ omap = out + OFF_MAP + (size_t)m * MAPELEMS;

    // A-matrix 16x4 f32 layout (ISA 7.12.2): lanes 0-15 hold K=0 (v0), K=1 (v1);
    // lanes 16-31 hold K=2 (v0), K=3 (v1). B 4x16 mirrors: N = lane&15,
    // K-halves split across lane halves.
    const int   kbase = (lane < 16) ? 0 : 2;
    const int   nlane = lane & 15;
    const float lmask = (lane < 16) ? 1.0f : 0.0f;  // mask duplicate M=8 row
    const float kf    = (float)kbase;

    float acc_z = 0.0f;   // sum |R|
    float acc_x = 0.0f;   // sum j*|R|
    float acc_y = 0.0f;   // sum i*|R|

    const v2f ones = {1.0f, 1.0f};

    for (int s = 0; s < 8; ++s) {
        const int col = s * 16 + nlane;
        // Per-slab lane base: all inner accesses become immediate offsets.
        const float* __restrict__ pb = p + kbase * WW + col;
        float* __restrict__ ob = omap + kbase * WW + col;

        v8f d0 = {0.f, 0.f, 0.f, 0.f, 0.f, 0.f, 0.f, 0.f};
        v8f d1 = {0.f, 0.f, 0.f, 0.f, 0.f, 0.f, 0.f, 0.f};

#pragma unroll
        for (int r0 = 0; r0 < HH; r0 += 4) {
            if ((r0 & 15) == 0 && r0 + 16 < HH) {
                __builtin_prefetch(&pb[(r0 + 16) * WW], 0, 3);
            }
            float bx = __builtin_fabsf(pb[r0 * WW]);
            float by = __builtin_fabsf(pb[r0 * WW + WW]);
            if (STORE_MAP) {                    // compile-time branch
                ob[r0 * WW]      = bx;
                ob[r0 * WW + WW] = by;
            }
            v2f bmat = {bx, by};
            // rows (r0+kbase, r0+kbase+1): r0 folds to an inline literal.
            v2f wrow = {kf + (float)r0, kf + (float)(r0 + 1)};
            // 8 args: (neg_a, A, neg_b, B, c_mod, C, reuse_a, reuse_b)
            d0 = __builtin_amdgcn_wmma_f32_16x16x4_f32(false, ones, false, bmat,
                                                       (short)0, d0, false, false);
            d1 = __builtin_amdgcn_wmma_f32_16x16x4_f32(false, wrow, false, bmat,
                                                       (short)0, d1, false, false);
        }

        // D row M=0 lives in v[0] of lanes 0-15 (N = lane); lanes 16-31 hold
        // the duplicate M=8 row. Mask via FMA (no exec manipulation).
        float c0 = d0[0] * lmask;
        float c1 = d1[0] * lmask;
        acc_z += c0;
        acc_x = __builtin_fmaf((float)col, c0, acc_x);
        acc_y += c1;
    }

    // wave32 butterfly reduction (lanes 16-31 contribute zeros)
    for (int off = 16; off >= 1; off >>= 1) {
        acc_z += __shfl_xor(acc_z, off, 32);
        acc_x += __shfl_xor(acc_x, off, 32);
        acc_y += __shfl_xor(acc_y, off, 32);
    }

    if (lane == 0) {
        float kx = rintf(acc_x / acc_z);   // RNE == jnp.round (half-to-even)
        float ky = rintf(acc_y / acc_z);
        if (STORE_MAP) {
            out[OFF_ZETA + m] = acc_z;
        }
        out[kp_off + 2 * m]     = kx;
        out[kp_off + 2 * m + 1] = ky;
    }
}

extern "C" void kernel_launch(void* const* d_in, const int* in_sizes, int n_in,
                              void* d_out, int out_size, void* d_ws, size_t ws_size,
                              hipStream_t stream) {
    const float* hm = (const float*)d_in[0];
    const float* tf = (const float*)d_in[1];
    float* out = (float*)d_out;
    // 1024 maps per tensor, 4 waves (one per map) per 128-thread block.
    dim3 grid(NMAPS / 4), block(128);
    hipLaunchKernelGGL((soft_argmax_wmma_kernel<true>), grid, block, 0, stream,
                       hm, out, OFF_KP);
    hipLaunchKernelGGL((soft_argmax_wmma_kernel<false>), grid, block, 0, stream,
                       tf, out, OFF_TFKP);
}